// VoronoiGridIntegrandSlang_5222680232007
// MI455X (gfx1250) — compile-verified
//
#include <hip/hip_runtime.h>

typedef float v2f __attribute__((ext_vector_type(2)));
typedef float v8f __attribute__((ext_vector_type(8)));

#define WAVES_PER_BLOCK 8
#define NSITES 64

__device__ __forceinline__ unsigned umin32(unsigned a, unsigned b) {
    return a < b ? a : b;
}

// D = A x B + C with V_WMMA_F32_16X16X4_F32.
// A (16x4)  : candidate rows  [-2px, -2py, |p|^2+2, 0]
// B (4x16)  : query columns   [qx, qy, 1, 0]^T
// D (16x16) : |p-q|^2 - |q|^2 + 4  (>0, argmin-equivalent to |p-q|^2 per column)
__global__ __launch_bounds__(256) void voronoi_wmma_kernel(
    const float* __restrict__ x, const float* __restrict__ p,
    float* __restrict__ out, int nq)
{
    const int lane = threadIdx.x & 31;
    const int half = lane >> 4;    // 0: lanes 0-15, 1: lanes 16-31
    const int mrow = lane & 15;
    const long long wave  = (long long)blockIdx.x * WAVES_PER_BLOCK + (threadIdx.x >> 5);
    const long long qbase = wave * 32;

    if (qbase + 32 > (long long)nq) {
        // Wave-uniform tail guard: scalar fallback keeps EXEC==all-ones off the WMMA path.
        long long q = qbase + lane;
        if (q < nq) {
            float qx = x[q * 2 + 0], qy = x[q * 2 + 1];
            float best = 1e30f; int bi = 0;
            for (int c = 0; c < NSITES; ++c) {
                float dx = qx - p[c * 5 + 0];
                float dy = qy - p[c * 5 + 1];
                float d2 = __builtin_fmaf(dx, dx, dy * dy);
                if (d2 < best) { best = d2; bi = c; }
            }
            out[q * 3 + 0] = p[bi * 5 + 2];
            out[q * 3 + 1] = p[bi * 5 + 3];
            out[q * 3 + 2] = p[bi * 5 + 4];
        }
        return;
    }

    // ---- A tiles (loop-invariant, 4 tiles x 2 VGPRs) -------------------------
    // f32 A 16x4 layout: lanes 0-15 hold K=0 (v.x), K=1 (v.y);
    //                    lanes 16-31 hold K=2 (v.x), K=3 (v.y).
    v2f A[4];
#pragma unroll
    for (int t = 0; t < 4; ++t) {
        const int c  = t * 16 + mrow;
        const float px = p[c * 5 + 0];
        const float py = p[c * 5 + 1];
        const float l2 = __builtin_fmaf(px, px, py * py) + 2.0f;
        A[t].x = half ? l2   : (-2.0f * px);
        A[t].y = half ? 0.0f : (-2.0f * py);
    }

    // ---- Queries: 32 per wave, coalesced 8B NT loads -------------------------
    const unsigned long long qraw =
        __builtin_nontemporal_load((const unsigned long long*)x + (qbase + lane));
    const float qx  = __uint_as_float((unsigned)(qraw & 0xFFFFFFFFull));
    const float qy  = __uint_as_float((unsigned)(qraw >> 32));
    const float qx2 = __shfl_xor(qx, 16, 32);   // other half's query
    const float qy2 = __shfl_xor(qy, 16, 32);

    // f32 B 4x16 layout: lanes 0-15 hold rows K=0 (v.x), K=1 (v.y);
    //                    lanes 16-31 hold rows K=2 (v.x), K=3 (v.y).
    v2f B0, B1;
    B0.x = half ? 1.0f : qx;    B0.y = half ? 0.0f : qy;    // queries qbase+0..15
    B1.x = half ? 1.0f : qx2;   B1.y = half ? 0.0f : qy2;   // queries qbase+16..31

    const v8f Cinit = {2.0f, 2.0f, 2.0f, 2.0f, 2.0f, 2.0f, 2.0f, 2.0f};

    // ---- 8 WMMAs: 4 candidate tiles x 2 query tiles --------------------------
    v8f D0[4], D1[4];
#pragma unroll
    for (int t = 0; t < 4; ++t) {
        D0[t] = __builtin_amdgcn_wmma_f32_16x16x4_f32(
            false, A[t], false, B0, (short)0, Cinit, false, false);
        D1[t] = __builtin_amdgcn_wmma_f32_16x16x4_f32(
            false, A[t], false, B1, (short)0, Cinit, false, false);
    }

    // ---- In-lane argmin via packed keys --------------------------------------
    // D values are > 0, so uint bit-pattern ordering == float ordering.
    // key = (bits(d) & ~63) | candidate_index; u32-min => argmin with
    // smallest-index tie-break on the truncated distance.
    // D tile row ownership: lane-half 0 -> rows 0-7 (VGPR v), half 1 -> rows 8-15.
    unsigned key0 = 0xFFFFFFFFu, key1 = 0xFFFFFFFFu;
#pragma unroll
    for (int t = 0; t < 4; ++t) {
#pragma unroll
        for (int v = 0; v < 8; ++v) {
            const unsigned idx = (unsigned)(t * 16 + half * 8 + v);
            const unsigned k0  = (__float_as_uint(D0[t][v]) & 0xFFFFFFC0u) | idx;
            const unsigned k1  = (__float_as_uint(D1[t][v]) & 0xFFFFFFC0u) | idx;
            key0 = umin32(key0, k0);
            key1 = umin32(key1, k1);
        }
    }
    // Combine the two row-halves of each column (lane n <-> lane n+16).
    key0 = umin32(key0, (unsigned)__shfl_xor((int)key0, 16, 32));
    key1 = umin32(key1, (unsigned)__shfl_xor((int)key1, 16, 32));

    // Lanes 0-15 emit queries qbase+0..15 (tile 0); lanes 16-31 emit
    // qbase+16..31 (tile 1) -> query id == qbase + lane for every lane.
    const unsigned key  = half ? key1 : key0;
    const unsigned widx = key & 63u;

    const float r = p[widx * 5 + 2];
    const float g = p[widx * 5 + 3];
    const float b = p[widx * 5 + 4];

    float* o = out + (qbase + lane) * 3;
    __builtin_nontemporal_store(r, o + 0);
    __builtin_nontemporal_store(g, o + 1);
    __builtin_nontemporal_store(b, o + 2);
}

extern "C" void kernel_launch(void* const* d_in, const int* in_sizes, int n_in,
                              void* d_out, int out_size, void* d_ws, size_t ws_size,
                              hipStream_t stream) {
    const float* x = (const float*)d_in[0];   // (N, 2) float32
    const float* p = (const float*)d_in[1];   // (64*5,) float32
    float* out = (float*)d_out;               // (N, 3) float32
    (void)d_ws; (void)ws_size; (void)n_in; (void)out_size;

    const int nq     = in_sizes[0] / 2;
    const int waves  = (nq + 31) / 32;
    const int blocks = (waves + WAVES_PER_BLOCK - 1) / WAVES_PER_BLOCK;

    voronoi_wmma_kernel<<<blocks, 256, 0, stream>>>(x, p, out, nq);
}